// IRLUT_70987219468576
// MI455X (gfx1250) — compile-verified
//
#include <hip/hip_runtime.h>
#include <hip/hip_bf16.h>

// ---------------------------------------------------------------------------
// CDNA5 (gfx1250) implementation of the IR-guided LUT enhancement pipeline.
// Matrix work (hp convs as implicit GEMM, both attention einsums) runs on
// V_WMMA_F32_16X16X4_F32 (full fp32: workload is bandwidth-bound, so fp32
// WMMA keeps reference numerics while using the matrix pipe).
// ---------------------------------------------------------------------------

typedef __attribute__((ext_vector_type(2))) float v2f;
typedef __attribute__((ext_vector_type(8))) float v8f;

// Only check / use the AMDGCN builtin during the device compilation pass;
// the host pass merely needs the function to parse.
#if defined(__HIP_DEVICE_COMPILE__)
#if !__has_builtin(__builtin_amdgcn_wmma_f32_16x16x4_f32)
#error "__builtin_amdgcn_wmma_f32_16x16x4_f32 not available on device toolchain"
#endif
#endif

__device__ __forceinline__ v8f wmma4(v2f a, v2f b, v8f c) {
  // D = A(16x4) * B(4x16) + C(16x16), all f32.
  // A frag: lane L<16 -> M=L, {K0,K1}; lane>=16 -> M=L-16, {K2,K3}
  // B frag: lane L<16 -> N=L, {K0,K1}; lane>=16 -> N=L-16, {K2,K3}
  // C/D   : VGPR i, lane L<16 -> [M=i][N=L]; lane>=16 -> [M=i+8][N=L-16]
#if defined(__HIP_DEVICE_COMPILE__)
  return __builtin_amdgcn_wmma_f32_16x16x4_f32(false, a, false, b, (short)0, c,
                                               false, false);
#else
  (void)a; (void)b;
  return c;  // host pass: parse-only stub, never executed
#endif
}

__device__ __forceinline__ float leaky(float x) { return x >= 0.f ? x : 0.05f * x; }

constexpr int   BN   = 2;
constexpr int   H0   = 448, W0 = 720;
constexpr int   HW0  = H0 * W0;               // 322560
constexpr int   H1   = 224, W1 = 360;
constexpr int   HW1  = H1 * W1;               // 80640
constexpr float EPSN = 1e-5f;

// ---------------------------------------------------------------------------
// 1) Patch-embed 3x3 conv (3->16) fused with LayerNorm over channels.
// One thread per pixel computes all 16 channels in registers.
// ---------------------------------------------------------------------------
__global__ void pe_ln_k(const float* __restrict__ img, const float* __restrict__ w,
                        const float* __restrict__ bias, const float* __restrict__ g,
                        const float* __restrict__ beta, float* __restrict__ out) {
  int tid = blockIdx.x * blockDim.x + threadIdx.x;
  if (tid >= BN * HW0) return;
  int pix = tid % HW0, b = tid / HW0;
  int h = pix / W0, x = pix % W0;
  float in[27];
#pragma unroll
  for (int c = 0; c < 3; ++c)
#pragma unroll
    for (int kh = 0; kh < 3; ++kh)
#pragma unroll
      for (int kw = 0; kw < 3; ++kw) {
        int hh = h + kh - 1, ww = x + kw - 1;
        in[c * 9 + kh * 3 + kw] =
            (hh >= 0 && hh < H0 && ww >= 0 && ww < W0)
                ? img[((size_t)(b * 3 + c) * H0 + hh) * W0 + ww] : 0.f;
      }
  float acc[16], mu = 0.f;
#pragma unroll
  for (int oc = 0; oc < 16; ++oc) {
    float s = bias[oc];
    const float* wk = w + oc * 27;
#pragma unroll
    for (int j = 0; j < 27; ++j) s += in[j] * wk[j];
    acc[oc] = s; mu += s;
  }
  mu *= (1.f / 16.f);
  float var = 0.f;
#pragma unroll
  for (int oc = 0; oc < 16; ++oc) { float d = acc[oc] - mu; var += d * d; }
  var *= (1.f / 16.f);
  float inv = rsqrtf(var + EPSN);
#pragma unroll
  for (int oc = 0; oc < 16; ++oc)
    out[((size_t)(b * 16 + oc) * H0 + h) * W0 + x] =
        (acc[oc] - mu) * inv * g[oc] + beta[oc];
}

// ---------------------------------------------------------------------------
// 2) hp conv: pixel_unshuffle(2) folded into implicit-GEMM WMMA conv.
// GEMM: M=16 out-channel tile (2 tiles for 32), N=16 pixel tile, K=576=64*9.
// One wave per (batch, m-tile, row, col-tile). Leaky applied at store.
// ---------------------------------------------------------------------------
__global__ void hp_conv_wmma_k(const float* __restrict__ xln,
                               const float* __restrict__ w,
                               const float* __restrict__ bias,
                               float* __restrict__ out) {
  int tx = blockIdx.x, h = blockIdx.y, bz = blockIdx.z;
  int b = bz >> 1, m0 = (bz & 1) * 16;
  int lane = threadIdx.x;
  int ml = lane & 15, kh2 = lane >> 4;
  int w0 = tx * 16;
  const float* xb = xln + (size_t)b * 16 * HW0;
  v8f acc = {0.f, 0.f, 0.f, 0.f, 0.f, 0.f, 0.f, 0.f};
  for (int k0 = 0; k0 < 576; k0 += 4) {
    int ka = k0 + 2 * kh2;
    v2f a;
    a.x = w[(size_t)(m0 + ml) * 576 + ka];
    a.y = w[(size_t)(m0 + ml) * 576 + ka + 1];
    float bv[2];
#pragma unroll
    for (int j = 0; j < 2; ++j) {
      int k = ka + j;
      int c64 = k / 9, rem = k - c64 * 9;
      int kh = rem / 3, kw = rem - kh * 3;
      int c16 = c64 >> 2, dy = (c64 >> 1) & 1, dx = c64 & 1;
      int hin = h + kh - 1, win = w0 + ml + kw - 1;
      float v = 0.f;
      if (hin >= 0 && hin < H1 && win >= 0 && win < W1)
        v = xb[((size_t)c16 * H0 + (2 * hin + dy)) * W0 + (2 * win + dx)];
      bv[j] = v;
    }
    v2f bb; bb.x = bv[0]; bb.y = bv[1];
    acc = wmma4(a, bb, acc);
  }
  int ww = w0 + ml;
  if (ww < W1) {
#pragma unroll
    for (int i = 0; i < 8; ++i) {
      int m = m0 + i + 8 * kh2;
      out[((size_t)(b * 32 + m) * H1 + h) * W1 + ww] = leaky(acc[i] + bias[m]);
    }
  }
}

// ---------------------------------------------------------------------------
// Row sum / sum-of-squares reduction (one block per row of length L).
// ---------------------------------------------------------------------------
__global__ void rowsumsq_k(const float* __restrict__ X, int L,
                           float* __restrict__ sums, float* __restrict__ ssq) {
  int row = blockIdx.x;
  const float* p = X + (size_t)row * L;
  float s = 0.f, s2 = 0.f;
  for (int i = threadIdx.x; i < L; i += blockDim.x) { float v = p[i]; s += v; s2 += v * v; }
  __shared__ float shs[256], shq[256];
  shs[threadIdx.x] = s; shq[threadIdx.x] = s2;
  __syncthreads();
  for (int off = 128; off > 0; off >>= 1) {
    if (threadIdx.x < off) {
      shs[threadIdx.x] += shs[threadIdx.x + off];
      shq[threadIdx.x] += shq[threadIdx.x + off];
    }
    __syncthreads();
  }
  if (threadIdx.x == 0) { sums[row] = shs[0]; ssq[row] = shq[0]; }
}

// Instance-norm apply (hp tensors): row = (b,c), C channels per batch.
__global__ void in_apply_k(float* __restrict__ X, const float* __restrict__ sums,
                           const float* __restrict__ ssq, const float* __restrict__ g,
                           const float* __restrict__ beta, int C) {
  size_t idx = (size_t)blockIdx.x * blockDim.x + threadIdx.x;
  size_t total = (size_t)BN * C * HW1;
  if (idx >= total) return;
  size_t row = idx / HW1;
  int c = (int)(row % C);
  float L = (float)HW1;
  float mu = sums[row] / L, var = ssq[row] / L - mu * mu;
  float a = g[c] * rsqrtf(var + EPSN);
  float cc = beta[c] - mu * a;
  X[idx] = a * X[idx] + cc;
}

// ---------------------------------------------------------------------------
// 3) Depthwise-grouped QKV conv (groups=32, 12 outs/group), leaky, scattered
// straight into the (B, head*16+row, HW) attention layout (pre-IN).
// ---------------------------------------------------------------------------
__global__ void qkv_conv_k(const float* __restrict__ Z, const float* __restrict__ w,
                           const float* __restrict__ bias, float* __restrict__ Qd,
                           float* __restrict__ Kd, float* __restrict__ Vd,
                           int e_start, int e_count) {
  size_t tid = (size_t)blockIdx.x * blockDim.x + threadIdx.x;
  size_t total = (size_t)BN * 32 * e_count * HW1;
  if (tid >= total) return;
  int pix = (int)(tid % HW1);
  size_t rest = tid / HW1;
  int je = (int)(rest % e_count); rest /= e_count;
  int c = (int)(rest % 32), b = (int)(rest / 32);
  int e = e_start + je, oc = c * 12 + e;
  int h = pix / W1, x = pix % W1;
  const float* src = Z + ((size_t)b * 32 + c) * HW1;
  const float* wk = w + oc * 9;
  float s = bias[oc];
#pragma unroll
  for (int kh = 0; kh < 3; ++kh) {
    int hh = h + kh - 1;
    if (hh < 0 || hh >= H1) continue;
#pragma unroll
    for (int kw = 0; kw < 3; ++kw) {
      int ww = x + kw - 1;
      if (ww < 0 || ww >= W1) continue;
      s += src[(size_t)hh * W1 + ww] * wk[kh * 3 + kw];
    }
  }
  s = leaky(s);
  float* dst; int rr;
  if (e < 4)      { dst = Qd; rr = e * 32 + c; }
  else if (e < 8) { dst = Kd; rr = (e - 4) * 32 + c; }
  else            { dst = Vd; rr = (e - 8) * 32 + c; }
  dst[((size_t)b * 128 + rr) * HW1 + pix] = s;
}

// IN apply on (B,128,HW) rows + optional fused L2 (F.normalize) scale.
// For y = a*x + cc:  ||y||^2 = a^2*ssq + 2*a*cc*sum + N*cc^2  (one pass).
__global__ void qkv_apply_k(float* __restrict__ X, const float* __restrict__ sums,
                            const float* __restrict__ ssq, const float* __restrict__ g,
                            const float* __restrict__ beta, int e_off, int do_l2) {
  size_t idx = (size_t)blockIdx.x * blockDim.x + threadIdx.x;
  size_t total = (size_t)BN * 128 * HW1;
  if (idx >= total) return;
  size_t row = idx / HW1;
  int ridx = (int)(row % 128);
  int e = (ridx >> 5) + e_off, c = ridx & 31, oc = c * 12 + e;
  float L = (float)HW1;
  float mu = sums[row] / L, var = ssq[row] / L - mu * mu;
  float a = g[oc] * rsqrtf(var + EPSN);
  float cc = beta[oc] - mu * a;
  float s = 1.f;
  if (do_l2) {
    float n2 = a * a * ssq[row] + 2.f * a * cc * sums[row] + L * cc * cc;
    s = 1.f / fmaxf(sqrtf(fmaxf(n2, 0.f)), 1e-12f);
  }
  X[idx] = (a * X[idx] + cc) * s;
}

// ---------------------------------------------------------------------------
// 4) Attention S = Q * K^T (16x16, K=HW1) with WMMA; K split over 16 waves,
// cross-wave LDS reduction. One workgroup per (b,head).
// ---------------------------------------------------------------------------
__global__ void attn_qk_k(const float* __restrict__ Q, const float* __restrict__ K,
                          float* __restrict__ S) {
  int bh = blockIdx.x;
  int lane = threadIdx.x & 31, wave = threadIdx.x >> 5;
  int ml = lane & 15, kh2 = lane >> 4;
  const float* qrow = Q + (size_t)bh * 16 * HW1 + (size_t)ml * HW1 + 2 * kh2;
  const float* krow = K + (size_t)bh * 16 * HW1 + (size_t)ml * HW1 + 2 * kh2;
  v8f acc = {0.f, 0.f, 0.f, 0.f, 0.f, 0.f, 0.f, 0.f};
  const int nchunk = HW1 / 4;  // 20160, exact
  for (int ch = wave; ch < nchunk; ch += 16) {
    int x0 = ch * 4;
    v2f a; a.x = qrow[x0]; a.y = qrow[x0 + 1];
    v2f b; b.x = krow[x0]; b.y = krow[x0 + 1];
    acc = wmma4(a, b, acc);
  }
  __shared__ float sh[16 * 32 * 8];  // 16 KB
#pragma unroll
  for (int i = 0; i < 8; ++i) sh[(wave * 32 + lane) * 8 + i] = acc[i];
  __syncthreads();
  if (wave == 0) {
#pragma unroll
    for (int i = 0; i < 8; ++i) {
      float s = 0.f;
      for (int wv = 0; wv < 16; ++wv) s += sh[(wv * 32 + lane) * 8 + i];
      int r = i + 8 * kh2;
      S[(size_t)bh * 256 + r * 16 + ml] = s;
    }
  }
}

// Scale by t[head] (and optional per-row 1/||q|| from sumsq) + row softmax.
__global__ void softmax16_k(float* __restrict__ S, const float* __restrict__ t,
                            const float* __restrict__ qssq) {
  int rid = blockIdx.x * blockDim.x + threadIdx.x;
  if (rid >= BN * 8 * 16) return;
  int hd = (rid >> 4) & 7;
  float sc = t[hd];
  if (qssq) sc *= 1.f / fmaxf(sqrtf(fmaxf(qssq[rid], 0.f)), 1e-12f);
  float* row = S + (size_t)rid * 16;
  float v[16], m = -1e30f;
#pragma unroll
  for (int j = 0; j < 16; ++j) { v[j] = row[j] * sc; m = fmaxf(m, v[j]); }
  float s = 0.f;
#pragma unroll
  for (int j = 0; j < 16; ++j) { v[j] = __expf(v[j] - m); s += v[j]; }
  float inv = 1.f / s;
#pragma unroll
  for (int j = 0; j < 16; ++j) row[j] = v[j] * inv;
}

// ---------------------------------------------------------------------------
// 5) O = A(16x16) @ V(16xHW1) with WMMA; N tiled by 16, A cached in LDS.
// ---------------------------------------------------------------------------
__global__ void attn_av_k(const float* __restrict__ A, const float* __restrict__ V,
                          float* __restrict__ O) {
  int bh = blockIdx.x;
  __shared__ float As[256];
  As[threadIdx.x] = A[(size_t)bh * 256 + threadIdx.x];
  As[threadIdx.x + 128] = A[(size_t)bh * 256 + threadIdx.x + 128];
  __syncthreads();
  int lane = threadIdx.x & 31, wave = threadIdx.x >> 5;
  int n = lane & 15, kh2 = lane >> 4;
  const float* Vb = V + (size_t)bh * 16 * HW1;
  float* Ob = O + (size_t)bh * 16 * HW1;
  const int ntiles = HW1 / 16;  // 5040, exact
  const int wpb = blockDim.x >> 5;
  for (int tl = blockIdx.y * wpb + wave; tl < ntiles; tl += gridDim.y * wpb) {
    int xt = tl * 16;
    v8f acc = {0.f, 0.f, 0.f, 0.f, 0.f, 0.f, 0.f, 0.f};
#pragma unroll
    for (int k0 = 0; k0 < 16; k0 += 4) {
      int kk = k0 + 2 * kh2;
      v2f a; a.x = As[n * 16 + kk]; a.y = As[n * 16 + kk + 1];
      v2f b; b.x = Vb[(size_t)kk * HW1 + xt + n];
             b.y = Vb[(size_t)(kk + 1) * HW1 + xt + n];
      acc = wmma4(a, b, acc);
    }
#pragma unroll
    for (int i = 0; i < 8; ++i) {
      int r = i + 8 * kh2;
      Ob[(size_t)r * HW1 + xt + n] = acc[i];
    }
  }
}

// ---------------------------------------------------------------------------
// 6) Fusion conv (groups=32: 4 in -> 1 out), reading O2 in attention layout
// via channel remap ch -> row (ch%8)*16 + ch/8 (the transpose is never
// materialized).
// ---------------------------------------------------------------------------
__global__ void fus_conv_k(const float* __restrict__ O2, const float* __restrict__ w,
                           const float* __restrict__ bias, float* __restrict__ feat) {
  size_t idx = (size_t)blockIdx.x * blockDim.x + threadIdx.x;
  size_t total = (size_t)BN * 32 * HW1;
  if (idx >= total) return;
  int pix = (int)(idx % HW1);
  int m = (int)((idx / HW1) % 32);
  int b = (int)(idx / ((size_t)32 * HW1));
  int h = pix / W1, x = pix % W1;
  float s = bias[m];
#pragma unroll
  for (int ci = 0; ci < 4; ++ci) {
    int ch = 4 * m + ci;
    int rr = (ch & 7) * 16 + (ch >> 3);
    const float* src = O2 + ((size_t)b * 128 + rr) * HW1;
    const float* wk = w + (size_t)(m * 4 + ci) * 9;
#pragma unroll
    for (int kh = 0; kh < 3; ++kh) {
      int hh = h + kh - 1;
      if (hh < 0 || hh >= H1) continue;
#pragma unroll
      for (int kw = 0; kw < 3; ++kw) {
        int ww = x + kw - 1;
        if (ww < 0 || ww >= W1) continue;
        s += src[(size_t)hh * W1 + ww] * wk[kh * 3 + kw];
      }
    }
  }
  feat[idx] = leaky(s);
}

// AdaptiveAvgPool2d(2): one block per (b, c, py, px) quadrant.
__global__ void pool2_k(const float* __restrict__ feat, float* __restrict__ p) {
  int blk = blockIdx.x;
  int px = blk & 1, py = (blk >> 1) & 1, cm = blk >> 2;  // cm = b*32+c
  const float* src = feat + (size_t)cm * HW1;
  int h0 = py * 112, w0 = px * 180;
  float s = 0.f;
  for (int i = threadIdx.x; i < 112 * 180; i += blockDim.x) {
    int hh = i / 180, ww = i % 180;
    s += src[(size_t)(h0 + hh) * W1 + w0 + ww];
  }
  __shared__ float sh[256];
  sh[threadIdx.x] = s;
  __syncthreads();
  for (int off = 128; off > 0; off >>= 1) {
    if (threadIdx.x < off) sh[threadIdx.x] += sh[threadIdx.x + off];
    __syncthreads();
  }
  if (threadIdx.x == 0) p[cm * 4 + py * 2 + px] = sh[0] / (112.f * 180.f);
}

// Tail stride-2 conv (2x2 -> 1x1, only kh,kw in {1,2} hit valid taps),
// leaky, then the 32->3 linear mixing head.
__global__ void tail_k(const float* __restrict__ p, const float* __restrict__ tw,
                       const float* __restrict__ tb, const float* __restrict__ lw,
                       const float* __restrict__ lb, float* __restrict__ wmix) {
  __shared__ float ps[128], p2[32];
  int b = blockIdx.x, m = threadIdx.x;
  for (int i = m; i < 128; i += 32) ps[i] = p[b * 128 + i];
  __syncthreads();
  float s = tb[m];
  for (int c = 0; c < 32; ++c)
#pragma unroll
    for (int kh = 1; kh < 3; ++kh)
#pragma unroll
      for (int kw = 1; kw < 3; ++kw)
        s += ps[c * 4 + (kh - 1) * 2 + (kw - 1)] * tw[((m * 32 + c) * 3 + kh) * 3 + kw];
  p2[m] = leaky(s);
  __syncthreads();
  if (m < 3) {
    float acc = lb[m];
    for (int c = 0; c < 32; ++c) acc += p2[c] * lw[m * 32 + c];
    wmix[b * 3 + m] = acc;
  }
}

// Per-sample combined LUT: lut_b = w0*lut0 + w1*lut1 + w2*lut2.
__global__ void lut_comb_k(const float* __restrict__ l0, const float* __restrict__ l1,
                           const float* __restrict__ l2, const float* __restrict__ wmix,
                           float* __restrict__ lutb) {
  const int N3 = 3 * 35937;
  size_t idx = (size_t)blockIdx.x * blockDim.x + threadIdx.x;
  if (idx >= (size_t)BN * N3) return;
  int i = (int)(idx % N3), b = (int)(idx / N3);
  lutb[idx] = wmix[b * 3 + 0] * l0[i] + wmix[b * 3 + 1] * l1[i] + wmix[b * 3 + 2] * l2[i];
}

// Trilinear 3D-LUT apply on the original image.
__global__ void trilinear_k(const float* __restrict__ img, const float* __restrict__ lutb,
                            float* __restrict__ out) {
  int tid = blockIdx.x * blockDim.x + threadIdx.x;
  if (tid >= BN * HW0) return;
  int pix = tid % HW0, b = tid / HW0;
  const float* ib = img + (size_t)b * 3 * HW0;
  const float binsize = 1.000001f / 32.f;
  float r = fminf(fmaxf(ib[pix], 0.f), 1.f);
  float g = fminf(fmaxf(ib[HW0 + pix], 0.f), 1.f);
  float bl = fminf(fmaxf(ib[2 * HW0 + pix], 0.f), 1.f);
  float fr = r / binsize, fg = g / binsize, fb = bl / binsize;
  int rid = min(max((int)fr, 0), 31);
  int gid = min(max((int)fg, 0), 31);
  int bid = min(max((int)fb, 0), 31);
  float rd = fr - rid, gd = fg - gid, bd = fb - bid;
  float w000 = (1 - rd) * (1 - gd) * (1 - bd), w001 = rd * (1 - gd) * (1 - bd);
  float w010 = (1 - rd) * gd * (1 - bd),       w011 = rd * gd * (1 - bd);
  float w100 = (1 - rd) * (1 - gd) * bd,       w101 = rd * (1 - gd) * bd;
  float w110 = (1 - rd) * gd * bd,             w111 = rd * gd * bd;
  const float* L = lutb + (size_t)b * 3 * 35937;
  int base = bid * 1089 + gid * 33 + rid;
#pragma unroll
  for (int c = 0; c < 3; ++c) {
    const float* Lc = L + c * 35937;
    float v = Lc[base] * w000 + Lc[base + 1] * w001 + Lc[base + 33] * w010 +
              Lc[base + 34] * w011 + Lc[base + 1089] * w100 + Lc[base + 1090] * w101 +
              Lc[base + 1122] * w110 + Lc[base + 1123] * w111;
    out[(size_t)(b * 3 + c) * HW0 + pix] = v;
  }
}

// ---------------------------------------------------------------------------
// Host launcher
// ---------------------------------------------------------------------------
static inline int nblk(size_t n, int t) { return (int)((n + (size_t)t - 1) / t); }

extern "C" void kernel_launch(void* const* d_in, const int* in_sizes, int n_in,
                              void* d_out, int out_size, void* d_ws, size_t ws_size,
                              hipStream_t stream) {
  (void)in_sizes; (void)n_in; (void)out_size; (void)ws_size;
  const float* img   = (const float*)d_in[0];
  const float* ir    = (const float*)d_in[1];
  const float* pe_w  = (const float*)d_in[2];
  const float* pe_b  = (const float*)d_in[3];
  const float* pie_w = (const float*)d_in[4];
  const float* pie_b = (const float*)d_in[5];
  const float* ln1_g = (const float*)d_in[6];
  const float* ln1_b = (const float*)d_in[7];
  const float* ln2_g = (const float*)d_in[8];
  const float* ln2_b = (const float*)d_in[9];
  const float* hp0_w = (const float*)d_in[10];
  const float* hp0_b = (const float*)d_in[11];
  const float* hp0_ig = (const float*)d_in[12];
  const float* hp0_ib = (const float*)d_in[13];
  const float* hp1_w = (const float*)d_in[14];
  const float* hp1_b = (const float*)d_in[15];
  const float* hp1_ig = (const float*)d_in[16];
  const float* hp1_ib = (const float*)d_in[17];
  const float* qkv_w = (const float*)d_in[18];
  const float* qkv_b = (const float*)d_in[19];
  const float* qkv_ig = (const float*)d_in[20];
  const float* qkv_ib = (const float*)d_in[21];
  const float* tt    = (const float*)d_in[22];
  const float* tt2   = (const float*)d_in[23];
  const float* fus_w = (const float*)d_in[24];
  const float* fus_b = (const float*)d_in[25];
  const float* tail_w = (const float*)d_in[26];
  const float* tail_b = (const float*)d_in[27];
  const float* lin_w = (const float*)d_in[28];
  const float* lin_b = (const float*)d_in[29];
  const float* lut0  = (const float*)d_in[30];
  const float* lut1  = (const float*)d_in[31];
  const float* lut2  = (const float*)d_in[32];

  // Workspace layout (floats). O aliases Q, O2 aliases K, feat aliases HX.
  const size_t SZ_XLN = (size_t)BN * 16 * HW0;   // 10,321,920
  const size_t SZ_HX  = (size_t)BN * 32 * HW1;   //  5,160,960
  const size_t SZ_QKV = (size_t)BN * 128 * HW1;  // 20,643,840
  float* ws   = (float*)d_ws;
  float* xln  = ws;
  float* iln  = xln + SZ_XLN;
  float* hx   = iln + SZ_XLN;
  float* hi   = hx + SZ_HX;
  float* Q    = hi + SZ_HX;
  float* Kx   = Q + SZ_QKV;
  float* Vx   = Kx + SZ_QKV;
  float* Kir  = Vx + SZ_QKV;
  float* Vir  = Kir + SZ_QKV;
  float* S    = Vir + SZ_QKV;      // 4096
  float* S2   = S + 4096;          // 4096
  float* sums = S2 + 4096;         // 256
  float* ssq  = sums + 256;        // 256
  float* pbuf = ssq + 256;         // 256
  float* wmix = pbuf + 256;        // 16
  float* lutb = wmix + 16;         // 2*3*35937
  float* O    = Q;                 // alias: Q dead after S
  float* O2   = Kx;                // alias: Kx dead after S
  float* feat = hx;                // alias: hx dead after qkv conv
  float* outp = (float*)d_out;

  // 1) patch embed + channel LN
  pe_ln_k<<<nblk((size_t)BN * HW0, 256), 256, 0, stream>>>(img, pe_w, pe_b, ln1_g, ln1_b, xln);
  pe_ln_k<<<nblk((size_t)BN * HW0, 256), 256, 0, stream>>>(ir, pie_w, pie_b, ln2_g, ln2_b, iln);

  // 2) hp convs (WMMA implicit GEMM) + instance norm
  dim3 hpG((W1 + 15) / 16, H1, BN * 2);
  hp_conv_wmma_k<<<hpG, 32, 0, stream>>>(xln, hp0_w, hp0_b, hx);
  hp_conv_wmma_k<<<hpG, 32, 0, stream>>>(iln, hp1_w, hp1_b, hi);
  rowsumsq_k<<<BN * 32, 256, 0, stream>>>(hx, HW1, sums, ssq);
  in_apply_k<<<nblk((size_t)BN * 32 * HW1, 256), 256, 0, stream>>>(hx, sums, ssq, hp0_ig, hp0_ib, 32);
  rowsumsq_k<<<BN * 32, 256, 0, stream>>>(hi, HW1, sums, ssq);
  in_apply_k<<<nblk((size_t)BN * 32 * HW1, 256), 256, 0, stream>>>(hi, sums, ssq, hp1_ig, hp1_ib, 32);

  // 3) qkv depthwise conv -> attention layout; then IN (+ fused L2 for q/k)
  qkv_conv_k<<<nblk((size_t)BN * 32 * 12 * HW1, 256), 256, 0, stream>>>(
      hx, qkv_w, qkv_b, Q, Kx, Vx, 0, 12);
  qkv_conv_k<<<nblk((size_t)BN * 32 * 8 * HW1, 256), 256, 0, stream>>>(
      hi, qkv_w, qkv_b, Kir, Kir, Vir, 4, 8);
  const int QB = nblk(SZ_QKV, 256);
  rowsumsq_k<<<BN * 128, 256, 0, stream>>>(Q, HW1, sums, ssq);
  qkv_apply_k<<<QB, 256, 0, stream>>>(Q, sums, ssq, qkv_ig, qkv_ib, 0, 1);
  rowsumsq_k<<<BN * 128, 256, 0, stream>>>(Kx, HW1, sums, ssq);
  qkv_apply_k<<<QB, 256, 0, stream>>>(Kx, sums, ssq, qkv_ig, qkv_ib, 4, 1);
  rowsumsq_k<<<BN * 128, 256, 0, stream>>>(Vx, HW1, sums, ssq);
  qkv_apply_k<<<QB, 256, 0, stream>>>(Vx, sums, ssq, qkv_ig, qkv_ib, 8, 0);
  rowsumsq_k<<<BN * 128, 256, 0, stream>>>(Kir, HW1, sums, ssq);
  qkv_apply_k<<<QB, 256, 0, stream>>>(Kir, sums, ssq, qkv_ig, qkv_ib, 4, 1);
  rowsumsq_k<<<BN * 128, 256, 0, stream>>>(Vir, HW1, sums, ssq);
  qkv_apply_k<<<QB, 256, 0, stream>>>(Vir, sums, ssq, qkv_ig, qkv_ib, 8, 0);

  // 4) attention 1: S = q k^T -> softmax(t) -> O = S v   (WMMA)
  attn_qk_k<<<BN * 8, 512, 0, stream>>>(Q, Kx, S);
  softmax16_k<<<1, 256, 0, stream>>>(S, tt, nullptr);
  attn_av_k<<<dim3(BN * 8, 80), 128, 0, stream>>>(S, Vx, O);

  // 5) attention 2: normalize O rows (fused via sumsq into softmax scale)
  rowsumsq_k<<<BN * 128, 256, 0, stream>>>(O, HW1, sums, ssq);
  attn_qk_k<<<BN * 8, 512, 0, stream>>>(O, Kir, S2);
  softmax16_k<<<1, 256, 0, stream>>>(S2, tt2, ssq);
  attn_av_k<<<dim3(BN * 8, 80), 128, 0, stream>>>(S2, Vir, O2);

  // 6) fusion conv -> pooled tail -> 3 LUT weights
  fus_conv_k<<<nblk((size_t)BN * 32 * HW1, 256), 256, 0, stream>>>(O2, fus_w, fus_b, feat);
  pool2_k<<<BN * 32 * 4, 256, 0, stream>>>(feat, pbuf);
  tail_k<<<BN, 32, 0, stream>>>(pbuf, tail_w, tail_b, lin_w, lin_b, wmix);

  // 7) combine LUTs per sample, trilinear apply
  lut_comb_k<<<nblk((size_t)BN * 3 * 35937, 256), 256, 0, stream>>>(lut0, lut1, lut2, wmix, lutb);
  trilinear_k<<<nblk((size_t)BN * HW0, 256), 256, 0, stream>>>(img, lutb, outp);
}